// CornerPooling_49452253446768
// MI455X (gfx1250) — compile-verified
//
#include <hip/hip_runtime.h>

#define NN 64
#define HH 512
#define WW 512
#define EPSV 1e-5f

// ---- CDNA5 async global->LDS copy (ASYNCcnt path) ------------------------
// ISA 15.18.3 opcode 96: GLOBAL_LOAD_ASYNC_TO_LDS_B32. VDST = VGPR holding
// the LDS byte offset, VADDR = 64-bit global address, tracked by ASYNCcnt.
__device__ __forceinline__ unsigned lds_off_of(const void* p) {
  // generic pointer to LDS: low 32 bits are the workgroup-relative LDS offset
  return (unsigned)(size_t)p;
}
__device__ __forceinline__ void async_ld_b32(unsigned lds_off, const float* gaddr) {
  asm volatile("global_load_async_to_lds_b32 %0, %1, off"
               :: "v"(lds_off), "v"(gaddr)
               : "memory");
}
__device__ __forceinline__ void async_wait0() {
  asm volatile("s_wait_asynccnt 0" ::: "memory");
}

// ==========================================================================
// Kernel 1: per-row  x1 = relu(bn(conv3x3(x)))  and  i1 = suffix-max over W.
// One 256-thread block per (n,h); 3 input rows staged in LDS via async loads.
// ==========================================================================
__global__ __launch_bounds__(256) void k_row(
    const float* __restrict__ x,
    const float* __restrict__ w3r, const float* __restrict__ b3r,
    const float* __restrict__ g3r, const float* __restrict__ be3r,
    const float* __restrict__ m3r, const float* __restrict__ v3r,
    float* __restrict__ x1, float* __restrict__ i1)
{
  __shared__ float row[3][WW + 4];   // cols -1..512 stored at +1
  __shared__ float scan[WW];
  const int tid = threadIdx.x;
  const int nh  = blockIdx.x;
  const int n   = nh >> 9;           // /512
  const int h   = nh & (HH - 1);
  const long long imgBase = (long long)n * HH * WW;

  #pragma unroll
  for (int r = 0; r < 3; ++r) {
    const int gh = h - 1 + r;
    const bool rowOK = ((unsigned)gh < (unsigned)HH);
    const float* src = x + imgBase + (long long)gh * WW;
    for (int j = tid; j < WW + 2; j += 256) {
      const int gw = j - 1;
      if (rowOK && (unsigned)gw < (unsigned)WW) {
        async_ld_b32(lds_off_of(&row[r][j]), src + gw);
      } else {
        row[r][j] = 0.0f;            // SAME zero padding
      }
    }
  }
  async_wait0();
  __syncthreads();

  // fold BN: bn(conv+b) = a*conv + (a*(b-m)+be)
  const float a    = g3r[0] * rsqrtf(v3r[0] + EPSV);
  const float bias = a * (b3r[0] - m3r[0]) + be3r[0];
  float k[9];
  #pragma unroll
  for (int i = 0; i < 9; ++i) k[i] = w3r[i];

  float* outRow = x1 + imgBase + (long long)h * WW;
  #pragma unroll
  for (int half = 0; half < 2; ++half) {
    const int w = tid + half * 256;
    float acc = 0.f;
    #pragma unroll
    for (int r = 0; r < 3; ++r)
      #pragma unroll
      for (int c = 0; c < 3; ++c)
        acc += k[r * 3 + c] * row[r][w + c];
    const float v = fmaxf(a * acc + bias, 0.0f);
    outRow[w] = v;
    scan[w]   = v;
  }
  __syncthreads();

  // inclusive right-to-left max scan (Hillis-Steele, 9 steps)
  const int w2 = tid + 256;
  #pragma unroll
  for (int off = 1; off < WW; off <<= 1) {
    const float t0 = (tid + off < WW) ? scan[tid + off] : -__builtin_inff();
    const float t1 = (w2  + off < WW) ? scan[w2  + off] : -__builtin_inff();
    __syncthreads();
    scan[tid] = fmaxf(scan[tid], t0);
    scan[w2]  = fmaxf(scan[w2],  t1);
    __syncthreads();
  }
  float* i1Row = i1 + imgBase + (long long)h * WW;
  i1Row[tid] = scan[tid];
  i1Row[w2]  = scan[w2];
}

// ==========================================================================
// Kernel 2: bottom-to-top column running max of x1, added in place into i1:
//           p[n,h,w] = i1[n,h,w] + max_{h'>=h} x1[n,h',w]
// ==========================================================================
#define PF 8   // prefetch distance (rows)
__global__ __launch_bounds__(64) void k_col(
    const float* __restrict__ x1, float* __restrict__ p)
{
  const int tid  = threadIdx.x;
  const int segs = WW / 64;                 // 8
  const int n    = blockIdx.x / segs;
  const int wseg = blockIdx.x % segs;
  const int w    = wseg * 64 + tid;
  const long long base = (long long)n * HH * WW + w;

  float running = -__builtin_inff();
  for (int h = HH - 1; h >= 0; --h) {
    const long long idx = base + (long long)h * WW;
    if (h >= PF) {                          // gfx1250 global_prefetch_b8
      __builtin_prefetch(&x1[idx - PF * WW], 0, 3);
      __builtin_prefetch(&p[idx - PF * WW], 0, 3);
    }
    running = fmaxf(running, x1[idx]);
    p[idx]  = p[idx] + running;
  }
}

// ==========================================================================
// Kernel 3: fully fused tail on 64x64 tiles (all intermediates in LDS):
//   s = relu(bn(conv3x3(p)) + bn(conv1x1(x)))
//   u = relu(bn(conv3x3(s)))   (shared weights)
//   v = relu(conv3x3(u) + b3rr)
//   out = w1*v + b1
// LDS: p tile 70x70 (halo 3), x/s tiles 68x68 (halo 2), u tile 66x66; ~78KB.
// ==========================================================================
#define TS  64
#define DP  (TS + 6)   // 70
#define DX  (TS + 4)   // 68
#define DU  (TS + 2)   // 66
#define LST 72         // LDS row stride
__global__ __launch_bounds__(256) void k_tail(
    const float* __restrict__ x, const float* __restrict__ p,
    const float* __restrict__ w3b, const float* __restrict__ b3b,
    const float* __restrict__ g3b, const float* __restrict__ be3b,
    const float* __restrict__ m3b, const float* __restrict__ v3b,
    const float* __restrict__ w1b, const float* __restrict__ b1b,
    const float* __restrict__ g1b, const float* __restrict__ be1b,
    const float* __restrict__ m1b, const float* __restrict__ v1b,
    const float* __restrict__ w3r, const float* __restrict__ b3r,
    const float* __restrict__ g3r, const float* __restrict__ be3r,
    const float* __restrict__ m3r, const float* __restrict__ v3r,
    const float* __restrict__ w3rr, const float* __restrict__ b3rr,
    const float* __restrict__ w1, const float* __restrict__ b1,
    float* __restrict__ out)
{
  __shared__ float sP[DP * LST];
  __shared__ float sX[DX * LST];
  __shared__ float sS[DX * LST];
  __shared__ float sU[DU * LST];
  const int tid = threadIdx.x;
  const int tilesPer = (HH / TS) * (WW / TS);     // 64
  const int n   = blockIdx.x / tilesPer;
  const int t   = blockIdx.x - n * tilesPer;
  const int th  = (t >> 3) * TS;                  // 8 tiles per row of tiles
  const int tw  = (t & 7) * TS;
  const long long imgBase = (long long)n * HH * WW;

  // p tile with halo 3 (70x70)
  for (int i = tid; i < DP * DP; i += 256) {
    const int ty = i / DP, tx = i - ty * DP;
    const int gh = th - 3 + ty, gw = tw - 3 + tx;
    if ((unsigned)gh < (unsigned)HH && (unsigned)gw < (unsigned)WW)
      async_ld_b32(lds_off_of(&sP[ty * LST + tx]),
                   p + imgBase + (long long)gh * WW + gw);
    else
      sP[ty * LST + tx] = 0.0f;
  }
  // x tile with halo 2 (68x68)
  for (int i = tid; i < DX * DX; i += 256) {
    const int ty = i / DX, tx = i - ty * DX;
    const int gh = th - 2 + ty, gw = tw - 2 + tx;
    if ((unsigned)gh < (unsigned)HH && (unsigned)gw < (unsigned)WW)
      async_ld_b32(lds_off_of(&sX[ty * LST + tx]),
                   x + imgBase + (long long)gh * WW + gw);
    else
      sX[ty * LST + tx] = 0.0f;
  }
  async_wait0();
  __syncthreads();

  const float a3b = g3b[0] * rsqrtf(v3b[0] + EPSV);
  const float c3b = a3b * (b3b[0] - m3b[0]) + be3b[0];
  const float a1b = g1b[0] * rsqrtf(v1b[0] + EPSV);
  const float kx  = a1b * w1b[0];
  const float cx  = a1b * (b1b[0] - m1b[0]) + be1b[0];
  const float a3r = g3r[0] * rsqrtf(v3r[0] + EPSV);
  const float c3r = a3r * (b3r[0] - m3r[0]) + be3r[0];
  float kb[9], kr[9], krr[9];
  #pragma unroll
  for (int i = 0; i < 9; ++i) { kb[i] = w3b[i]; kr[i] = w3r[i]; krr[i] = w3rr[i]; }
  const float w1c = w1[0], b1c = b1[0], brr = b3rr[0];

  // s tile (68x68); s == 0 outside the image (SAME zero padding of next conv)
  for (int i = tid; i < DX * DX; i += 256) {
    const int ty = i / DX, tx = i - ty * DX;
    const int gh = th - 2 + ty, gw = tw - 2 + tx;
    float s = 0.f;
    if ((unsigned)gh < (unsigned)HH && (unsigned)gw < (unsigned)WW) {
      float acc = 0.f;
      #pragma unroll
      for (int r = 0; r < 3; ++r)
        #pragma unroll
        for (int c = 0; c < 3; ++c)
          acc += kb[r * 3 + c] * sP[(ty + r) * LST + tx + c];
      s = fmaxf(a3b * acc + c3b + kx * sX[ty * LST + tx] + cx, 0.f);
    }
    sS[ty * LST + tx] = s;
  }
  __syncthreads();

  // u tile (66x66)
  for (int i = tid; i < DU * DU; i += 256) {
    const int ty = i / DU, tx = i - ty * DU;
    const int gh = th - 1 + ty, gw = tw - 1 + tx;
    float u = 0.f;
    if ((unsigned)gh < (unsigned)HH && (unsigned)gw < (unsigned)WW) {
      float acc = 0.f;
      #pragma unroll
      for (int r = 0; r < 3; ++r)
        #pragma unroll
        for (int c = 0; c < 3; ++c)
          acc += kr[r * 3 + c] * sS[(ty + r) * LST + tx + c];
      u = fmaxf(a3r * acc + c3r, 0.f);
    }
    sU[ty * LST + tx] = u;
  }
  __syncthreads();

  // out tile (64x64)
  for (int i = tid; i < TS * TS; i += 256) {
    const int ty = i >> 6, tx = i & 63;
    float acc = 0.f;
    #pragma unroll
    for (int r = 0; r < 3; ++r)
      #pragma unroll
      for (int c = 0; c < 3; ++c)
        acc += krr[r * 3 + c] * sU[(ty + r) * LST + tx + c];
    const float v = fmaxf(acc + brr, 0.f);
    out[imgBase + (long long)(th + ty) * WW + (tw + tx)] = w1c * v + b1c;
  }
}

extern "C" void kernel_launch(void* const* d_in, const int* in_sizes, int n_in,
                              void* d_out, int out_size, void* d_ws, size_t ws_size,
                              hipStream_t stream) {
  (void)in_sizes; (void)n_in; (void)out_size; (void)ws_size;
  const float* x    = (const float*)d_in[0];
  const float* w3r  = (const float*)d_in[1];
  const float* b3r  = (const float*)d_in[2];
  const float* g3r  = (const float*)d_in[3];
  const float* be3r = (const float*)d_in[4];
  const float* m3r  = (const float*)d_in[5];
  const float* v3r  = (const float*)d_in[6];
  const float* w3b  = (const float*)d_in[7];
  const float* b3b  = (const float*)d_in[8];
  const float* g3b  = (const float*)d_in[9];
  const float* be3b = (const float*)d_in[10];
  const float* m3b  = (const float*)d_in[11];
  const float* v3b  = (const float*)d_in[12];
  const float* w1b  = (const float*)d_in[13];
  const float* b1b  = (const float*)d_in[14];
  const float* g1b  = (const float*)d_in[15];
  const float* be1b = (const float*)d_in[16];
  const float* m1b  = (const float*)d_in[17];
  const float* v1b  = (const float*)d_in[18];
  const float* w3rr = (const float*)d_in[19];
  const float* b3rr = (const float*)d_in[20];
  const float* w1   = (const float*)d_in[21];
  const float* b1   = (const float*)d_in[22];
  float* out = (float*)d_out;

  float* ws0 = (float*)d_ws;                           // x1  (64 MiB)
  float* ws1 = ws0 + (size_t)NN * HH * WW;             // i1 -> p (64 MiB)

  k_row<<<NN * HH, 256, 0, stream>>>(x, w3r, b3r, g3r, be3r, m3r, v3r, ws0, ws1);
  k_col<<<NN * (WW / 64), 64, 0, stream>>>(ws0, ws1);
  k_tail<<<NN * ((HH / TS) * (WW / TS)), 256, 0, stream>>>(x, ws1,
      w3b, b3b, g3b, be3b, m3b, v3b,
      w1b, b1b, g1b, be1b, m1b, v1b,
      w3r, b3r, g3r, be3r, m3r, v3r,
      w3rr, b3rr, w1, b1, out);
}